// CRF_18545668784276
// MI455X (gfx1250) — compile-verified
//
#include <hip/hip_runtime.h>
#include <hip/hip_bf16.h>

// Problem constants from the reference.
#define BDIM 256
#define SDIM 2048
#define TDIM 64

#define LOG2E 1.44269504088896340736f
#define LN2   0.69314718055994530942f

typedef float v2f __attribute__((ext_vector_type(2)));
typedef float v8f __attribute__((ext_vector_type(8)));

// ---------------------------------------------------------------------------
// Kernel 1: expT[i][j] = exp(transitions[i][j])   (64x64)
// ---------------------------------------------------------------------------
__global__ __launch_bounds__(256) void crf_exptrans(const float* __restrict__ trans,
                                                    float* __restrict__ expT) {
    int i = blockIdx.x * 256 + threadIdx.x;
    if (i < TDIM * TDIM) expT[i] = __expf(trans[i]);
}

// ---------------------------------------------------------------------------
// Kernel 2: gold score per batch.
// gold[b] = em[b,0,tg0] + sum_{t=1..S-1} (trans[tg[t-1],tg[t]] + em[b,t,tg[t]]) * mask[b,t]
// ---------------------------------------------------------------------------
__global__ __launch_bounds__(256) void crf_gold(const float* __restrict__ em,
                                                const float* __restrict__ trans,
                                                const int* __restrict__ tags,
                                                const unsigned char* __restrict__ mask,
                                                float* __restrict__ gold) {
    int b = blockIdx.x;
    const int* tg = tags + (size_t)b * SDIM;
    const unsigned char* mk = mask + (size_t)b * SDIM;
    const float* e = em + (size_t)b * SDIM * TDIM;
    float acc = 0.0f;
    for (int t = 1 + threadIdx.x; t < SDIM; t += 256) {
        int tp = tg[t - 1];
        int tc = tg[t];
        float v = trans[tp * TDIM + tc] + e[(size_t)t * TDIM + tc];
        acc += mk[t] ? v : 0.0f;
    }
    __shared__ float red[256];
    red[threadIdx.x] = acc;
    __syncthreads();
    for (int s = 128; s > 0; s >>= 1) {
        if (threadIdx.x < (unsigned)s) red[threadIdx.x] += red[threadIdx.x + s];
        __syncthreads();
    }
    if (threadIdx.x == 0) gold[b] = red[0] + e[tg[0]];
}

// ---------------------------------------------------------------------------
// Kernel 3: forward scan (log-partition) via WMMA f32 16x16x4 in exp-space,
// carried in base-2 units: q = score * log2(e).
//   p = exp2(q - m)        (raw v_exp_f32)
//   v = p @ exp(trans)     (16x v_wmma_f32_16x16x4_f32, dual accumulators)
//   q' = m + log2(v) + e*log2(e)   (raw v_log_f32)
// Grid: 16 blocks (batch groups of 16) x 128 threads (4 waves; wave w owns
// state columns j in [16w, 16w+16)).
// C-tile layout (16x16 f32): lane 0-15 -> N=lane, VGPR r -> M=r ; lanes 16-31
// -> same N, M=r+8.  A (16x4) / B (4x16): VGPR v -> K = v + 2*(lane>=16).
// Steady-state loop is fully unconditional (last two steps peeled) and
// unrolled x3 so the 3-deep emission register pipeline rotates by renaming.
// ---------------------------------------------------------------------------
__global__ __launch_bounds__(128) void crf_forward_scan(const float* __restrict__ em,
                                                        const float* __restrict__ expT,
                                                        const unsigned char* __restrict__ mask,
                                                        float* __restrict__ logz) {
    __shared__ __align__(16) float pbuf[16 * 68];   // p matrix, padded stride 68 (bank-conflict free)
    __shared__ __align__(16) float wmax[16][4];     // per-wave partial row maxes
    __shared__ __align__(16) float wsum[16][4];     // per-wave partial row sums (final LSE)
    __shared__ __align__(16) float mrow_s[16];      // combined row max (final LSE)

    const int tid  = threadIdx.x;
    const int wv   = tid >> 5;      // wave id == N-tile id
    const int lane = tid & 31;
    const int hi   = lane >> 4;     // half-wave select
    const int ln   = lane & 15;
    const int bg   = blockIdx.x;    // batch group (16 batches)

    const int jcol = wv * 16 + ln;                       // this lane's state column
    const size_t bbase = (size_t)(bg * 16) * SDIM * TDIM;

    // ---- Preload this wave's B slice of expT: 16 K-chunks of (4 x 16). ----
    v2f Bt[16];
#pragma unroll
    for (int kc = 0; kc < 16; ++kc) {
        int i0 = 4 * kc + 2 * hi;
        Bt[kc].x = expT[i0 * TDIM + jcol];
        Bt[kc].y = expT[(i0 + 1) * TDIM + jcol];
    }

    // ---- score init in base-2 units: q = em[...,0,:] * log2(e) ----
    float q[8];
#pragma unroll
    for (int r = 0; r < 8; ++r) {
        int row = r + 8 * hi;
        q[r] = em[bbase + (size_t)row * SDIM * TDIM + jcol] * LOG2E;
    }

    // emission loader: e2 = em*log2e, mk = mask?1:0, mi = 1-mk (all off critical path)
    auto load_em = [&](int t, float (&e)[8], float (&mk)[8], float (&mi)[8]) {
#pragma unroll
        for (int r = 0; r < 8; ++r) {
            int row = r + 8 * hi;
            e[r] = em[bbase + (size_t)row * SDIM * TDIM + (size_t)t * TDIM + jcol] * LOG2E;
            float f = mask[(size_t)(bg * 16 + row) * SDIM + t] ? 1.0f : 0.0f;
            mk[r] = f;
            mi[r] = 1.0f - f;
        }
    };

    // one scan step (t-independent; emissions arrive via registers)
    auto one_step = [&](const float (&e)[8], const float (&mk)[8], const float (&mi)[8]) {
        // 1) per-wave row max over its 16 columns (butterfly within 16-lane half)
        float rm[8];
#pragma unroll
        for (int r = 0; r < 8; ++r) {
            float v = q[r];
            v = fmaxf(v, __shfl_xor(v, 1, 32));
            v = fmaxf(v, __shfl_xor(v, 2, 32));
            v = fmaxf(v, __shfl_xor(v, 4, 32));
            v = fmaxf(v, __shfl_xor(v, 8, 32));
            rm[r] = v;
        }
        if (ln == 0) {  // lanes 0 and 16 publish rows 0-7 / 8-15
#pragma unroll
            for (int r = 0; r < 8; ++r) wmax[r + 8 * hi][wv] = rm[r];
        }
        __syncthreads();

        // 2) combined row max across the 4 waves (broadcast b128 reads)
        float m[8];
#pragma unroll
        for (int r = 0; r < 8; ++r) {
            const float4 c4 = *(const float4*)&wmax[r + 8 * hi][0];
            m[r] = fmaxf(fmaxf(c4.x, c4.y), fmaxf(c4.z, c4.w));
        }

        // 3) p = exp2(q - m)  ->  LDS (row-major, stride 68)
#pragma unroll
        for (int r = 0; r < 8; ++r) {
            pbuf[(r + 8 * hi) * 68 + jcol] = __builtin_amdgcn_exp2f(q[r] - m[r]);
        }
        __syncthreads();

        // 4) v = p @ expT via 16x f32 WMMA, two independent accumulator chains
        v8f acc0 = {};
        v8f acc1 = {};
#pragma unroll
        for (int kc = 0; kc < 8; ++kc) {
            const v2f a0 = *(const v2f*)&pbuf[ln * 68 + 4 * (2 * kc) + 2 * hi];
            const v2f a1 = *(const v2f*)&pbuf[ln * 68 + 4 * (2 * kc + 1) + 2 * hi];
            acc0 = __builtin_amdgcn_wmma_f32_16x16x4_f32(
                false, a0, false, Bt[2 * kc], (short)0, acc0, false, false);
            acc1 = __builtin_amdgcn_wmma_f32_16x16x4_f32(
                false, a1, false, Bt[2 * kc + 1], (short)0, acc1, false, false);
        }

        // 5) q' = m + log2(v) + e2 ; branchless exact masked select
#pragma unroll
        for (int r = 0; r < 8; ++r) {
            float v  = acc0[r] + acc1[r];
            float nv = m[r] + __builtin_amdgcn_logf(v) + e[r];
            q[r] = mk[r] * nv + mi[r] * q[r];
        }
    };

    // ---- 3-deep emission/mask register pipeline ----
    float eC[8], eN1[8], eN2[8];
    float mC[8], mN1[8], mN2[8];
    float iC[8], iN1[8], iN2[8];
    load_em(1, eC, mC, iC);
    load_em(2, eN1, mN1, iN1);

    // steady state: everything unconditional; unroll x3 -> rotation by renaming
#pragma unroll 3
    for (int t = 1; t <= SDIM - 3; ++t) {
        load_em(t + 2, eN2, mN2, iN2);
        one_step(eC, mC, iC);
#pragma unroll
        for (int r = 0; r < 8; ++r) {
            eC[r] = eN1[r];  mC[r] = mN1[r];  iC[r] = iN1[r];
            eN1[r] = eN2[r]; mN1[r] = mN2[r]; iN1[r] = iN2[r];
        }
    }
    // peeled epilogue: t = SDIM-2 and t = SDIM-1
    one_step(eC, mC, iC);
    one_step(eN1, mN1, iN1);

    // ---- final logsumexp over all 64 states per row (still base-2) ----
    float rm[8];
#pragma unroll
    for (int r = 0; r < 8; ++r) {
        float v = q[r];
        v = fmaxf(v, __shfl_xor(v, 1, 32));
        v = fmaxf(v, __shfl_xor(v, 2, 32));
        v = fmaxf(v, __shfl_xor(v, 4, 32));
        v = fmaxf(v, __shfl_xor(v, 8, 32));
        rm[r] = v;
    }
    if (ln == 0) {
#pragma unroll
        for (int r = 0; r < 8; ++r) wmax[r + 8 * hi][wv] = rm[r];
    }
    __syncthreads();
    float m[8];
#pragma unroll
    for (int r = 0; r < 8; ++r) {
        const float4 c4 = *(const float4*)&wmax[r + 8 * hi][0];
        m[r] = fmaxf(fmaxf(c4.x, c4.y), fmaxf(c4.z, c4.w));
    }
#pragma unroll
    for (int r = 0; r < 8; ++r) {
        float v = __builtin_amdgcn_exp2f(q[r] - m[r]);
        v += __shfl_xor(v, 1, 32);
        v += __shfl_xor(v, 2, 32);
        v += __shfl_xor(v, 4, 32);
        v += __shfl_xor(v, 8, 32);
        rm[r] = v;  // per-wave partial sum for row r+8*hi
    }
    if (ln == 0) {
#pragma unroll
        for (int r = 0; r < 8; ++r) wsum[r + 8 * hi][wv] = rm[r];
        if (wv == 0) {
#pragma unroll
            for (int r = 0; r < 8; ++r) mrow_s[r + 8 * hi] = m[r];
        }
    }
    __syncthreads();
    if (tid < 16) {
        float tot = wsum[tid][0] + wsum[tid][1] + wsum[tid][2] + wsum[tid][3];
        // back to natural log units
        logz[bg * 16 + tid] = LN2 * (mrow_s[tid] + __builtin_amdgcn_logf(tot));
    }
}

// ---------------------------------------------------------------------------
// Kernel 4: out = -mean(gold - logz)
// ---------------------------------------------------------------------------
__global__ __launch_bounds__(256) void crf_finalize(const float* __restrict__ gold,
                                                    const float* __restrict__ logz,
                                                    float* __restrict__ out) {
    __shared__ float red[256];
    int b = threadIdx.x;
    red[b] = gold[b] - logz[b];
    __syncthreads();
    for (int s = 128; s > 0; s >>= 1) {
        if (b < s) red[b] += red[b + s];
        __syncthreads();
    }
    if (b == 0) out[0] = -red[0] / (float)BDIM;
}

// ---------------------------------------------------------------------------
// Launch. Inputs in setup_inputs() order: emissions, transitions, tags, mask.
// tags: JAX default x64-disabled -> int32 (bridge doc: integer -> const int*).
// mask: bool -> 1 byte per element.
// ws layout (floats): [0,4096) expT ; [4096,4352) gold ; [4352,4608) logz.
// ---------------------------------------------------------------------------
extern "C" void kernel_launch(void* const* d_in, const int* in_sizes, int n_in,
                              void* d_out, int out_size, void* d_ws, size_t ws_size,
                              hipStream_t stream) {
    const float* emissions    = (const float*)d_in[0];
    const float* transitions  = (const float*)d_in[1];
    const int* tags           = (const int*)d_in[2];
    const unsigned char* mask = (const unsigned char*)d_in[3];
    float* out = (float*)d_out;
    float* ws  = (float*)d_ws;

    float* expT = ws;
    float* gold = ws + 4096;
    float* logz = ws + 4096 + 256;

    crf_exptrans<<<dim3(16), dim3(256), 0, stream>>>(transitions, expT);
    crf_gold<<<dim3(BDIM), dim3(256), 0, stream>>>(emissions, transitions, tags, mask, gold);
    crf_forward_scan<<<dim3(BDIM / 16), dim3(128), 0, stream>>>(emissions, expT, mask, logz);
    crf_finalize<<<dim3(1), dim3(256), 0, stream>>>(gold, logz, out);
}